// Vanilla_GCN_9809705304380
// MI455X (gfx1250) — compile-verified
//
#include <hip/hip_runtime.h>
#include <math.h>

typedef float v2f __attribute__((ext_vector_type(2)));
typedef float v8f __attribute__((ext_vector_type(8)));

#define N_NODES 100000
#define N_EDGES 1600000
#define DIM_IN  128
#define DIM_H   64
#define DIM_C   40
#define LN_EPS  1e-5f

// ---------------------------------------------------------------------------
// utility: zero-fill
// ---------------------------------------------------------------------------
__global__ void fill_zero(float* __restrict__ p, long n) {
    long i = (long)blockIdx.x * blockDim.x + threadIdx.x;
    if (i < n) p[i] = 0.0f;
}

// ---------------------------------------------------------------------------
// in-degree of dst (as float; exact for counts < 2^24)
// ---------------------------------------------------------------------------
__global__ void degree_kernel(const int* __restrict__ dst, float* __restrict__ deg, int E) {
    int e = blockIdx.x * blockDim.x + threadIdx.x;
    if (e < E) unsafeAtomicAdd(&deg[dst[e]], 1.0f);
}

__global__ void dinv_kernel(float* __restrict__ deg, int n) {
    int i = blockIdx.x * blockDim.x + threadIdx.x;
    if (i < n) {
        float d = deg[i];
        deg[i] = (d > 0.0f) ? rsqrtf(d) : 0.0f;
    }
}

__global__ void norm_kernel(const int* __restrict__ src, const int* __restrict__ dst,
                            const float* __restrict__ dinv, float* __restrict__ norm, int E) {
    int e = blockIdx.x * blockDim.x + threadIdx.x;
    if (e < E) norm[e] = dinv[src[e]] * dinv[dst[e]];
}

// ---------------------------------------------------------------------------
// fp32 WMMA GEMM:  D[M x bw] = A[M x K] @ B[K x bw]
//   - one 16x16 output tile per wave (v_wmma_f32_16x16x4_f32, K stepped by 4)
//   - blockDim.x = 32 * tilesN ; wave w computes column tile w
//   - bw = B row-stride = D row-stride = valid-column count.
//     For padded column tiles (final layer: 40 valid of 48), the column index
//     is CLAMPED (not masked) for B loads: out-of-range B columns only feed
//     out-of-range D columns, which are skipped at store time. This keeps the
//     inner loop free of EXEC-mask branches (loads are unconditional) and
//     keeps EXEC all-1s for WMMA.
//   - gridDim.x = M/16 (M = 100000 is an exact multiple of 16)
// ---------------------------------------------------------------------------
__global__ void gemm16_wmma(const float* __restrict__ A, const float* __restrict__ B,
                            float* __restrict__ D, int K, int bw) {
    const int lane  = threadIdx.x & 31;
    const int tileN = threadIdx.x >> 5;
    const int tileM = blockIdx.x;
    const int half  = lane >> 4;     // 0: K={0,1}, 1: K={2,3}
    const int l16   = lane & 15;

    const long row  = (long)tileM * 16 + l16;   // A row (same for both half-waves)
    const int  col  = tileN * 16 + l16;         // B/D column
    const int  kb   = half * 2;
    const bool cok  = (col < bw);
    const int  ccol = cok ? col : (bw - 1);     // clamped, always in-bounds

    const float* __restrict__ arow = A + row * K;
    const float* __restrict__ bcol = B + kb * bw + ccol;

    v8f acc = {};
    for (int k = 0; k < K; k += 4) {
        v2f a, b;
        a.x = arow[k + kb + 0];
        a.y = arow[k + kb + 1];
        b.x = bcol[(long)k * bw];
        b.y = bcol[(long)k * bw + bw];
        // 8 args: (neg_a, A, neg_b, B, c_mod, C, reuse_a, reuse_b)
        acc = __builtin_amdgcn_wmma_f32_16x16x4_f32(false, a, false, b,
                                                    (short)0, acc, false, false);
    }

    if (cok) {
        const long rbase = (long)tileM * 16 + half * 8;
        #pragma unroll
        for (int r = 0; r < 8; ++r)
            D[(rbase + r) * bw + col] = acc[r];
    }
}

// ---------------------------------------------------------------------------
// edge scatter: agg[dst] += hw[src] * norm   (64 features, 16 lanes/edge x float4)
// ---------------------------------------------------------------------------
__global__ void scatter_kernel(const float* __restrict__ hw,
                               const int* __restrict__ src, const int* __restrict__ dst,
                               const float* __restrict__ norm,
                               float* __restrict__ agg, int E) {
    long tid = (long)blockIdx.x * blockDim.x + threadIdx.x;
    long e   = tid >> 4;
    if (e >= E) return;
    int f = (int)(tid & 15) * 4;

    int   s = src[e];
    int   d = dst[e];
    float w = norm[e];

    const float4 v = *(const float4*)(hw + (long)s * DIM_H + f);
    float* base = agg + (long)d * DIM_H + f;
    unsafeAtomicAdd(base + 0, v.x * w);
    unsafeAtomicAdd(base + 1, v.y * w);
    unsafeAtomicAdd(base + 2, v.z * w);
    unsafeAtomicAdd(base + 3, v.w * w);
}

// ---------------------------------------------------------------------------
// bias + LayerNorm(64) + ReLU : one wave32 per node, 2 features per lane
// ---------------------------------------------------------------------------
__global__ void ln_relu_kernel(const float* __restrict__ agg,
                               const float* __restrict__ bias,
                               const float* __restrict__ gamma,
                               const float* __restrict__ beta,
                               float* __restrict__ h, int n) {
    long gtid = (long)blockIdx.x * blockDim.x + threadIdx.x;
    int  node = (int)(gtid >> 5);
    int  lane = threadIdx.x & 31;
    if (node >= n) return;

    const float* row = agg + (long)node * DIM_H;
    float v0 = row[lane]      + bias[lane];
    float v1 = row[lane + 32] + bias[lane + 32];

    float s = v0 + v1;
    #pragma unroll
    for (int off = 16; off > 0; off >>= 1) s += __shfl_xor(s, off, 32);
    float mu = s * (1.0f / 64.0f);

    float d0 = v0 - mu, d1 = v1 - mu;
    float sq = d0 * d0 + d1 * d1;
    #pragma unroll
    for (int off = 16; off > 0; off >>= 1) sq += __shfl_xor(sq, off, 32);
    float inv = rsqrtf(sq * (1.0f / 64.0f) + LN_EPS);

    float o0 = gamma[lane]      * d0 * inv + beta[lane];
    float o1 = gamma[lane + 32] * d1 * inv + beta[lane + 32];

    h[(long)node * DIM_H + lane]      = fmaxf(o0, 0.0f);
    h[(long)node * DIM_H + lane + 32] = fmaxf(o1, 0.0f);
}

// ---------------------------------------------------------------------------
// add final bias + log_softmax over 40 classes, in place on d_out
// ---------------------------------------------------------------------------
__global__ void logsoftmax_kernel(float* __restrict__ out, const float* __restrict__ bf, int n) {
    long node = (long)blockIdx.x * blockDim.x + threadIdx.x;
    if (node >= n) return;
    float* row = out + node * DIM_C;

    float v[DIM_C];
    float mx = -INFINITY;
    #pragma unroll
    for (int i = 0; i < DIM_C; ++i) {
        v[i] = row[i] + bf[i];
        mx = fmaxf(mx, v[i]);
    }
    float se = 0.0f;
    #pragma unroll
    for (int i = 0; i < DIM_C; ++i) se += expf(v[i] - mx);
    float lse = mx + logf(se);
    #pragma unroll
    for (int i = 0; i < DIM_C; ++i) row[i] = v[i] - lse;
}

// ---------------------------------------------------------------------------
// launcher
// ---------------------------------------------------------------------------
extern "C" void kernel_launch(void* const* d_in, const int* in_sizes, int n_in,
                              void* d_out, int out_size, void* d_ws, size_t ws_size,
                              hipStream_t stream) {
    const float* x   = (const float*)d_in[0];
    const int*   ei  = (const int*)d_in[1];
    const int*   src = ei;
    const int*   dst = ei + N_EDGES;

    const float* W[3] = { (const float*)d_in[2],  (const float*)d_in[6],  (const float*)d_in[10] };
    const float* b[3] = { (const float*)d_in[3],  (const float*)d_in[7],  (const float*)d_in[11] };
    const float* g[3] = { (const float*)d_in[4],  (const float*)d_in[8],  (const float*)d_in[12] };
    const float* t[3] = { (const float*)d_in[5],  (const float*)d_in[9],  (const float*)d_in[13] };
    const float* Wf   = (const float*)d_in[14];
    const float* bf   = (const float*)d_in[15];
    float* out        = (float*)d_out;

    float* ws   = (float*)d_ws;
    float* dinv = ws;                           // N
    float* norm = dinv + N_NODES;               // E
    float* hw   = norm + N_EDGES;               // N*64
    float* agg  = hw   + (long)N_NODES * DIM_H; // N*64
    float* h    = agg  + (long)N_NODES * DIM_H; // N*64

    // --- GCN symmetric normalization ---
    fill_zero<<<(N_NODES + 255) / 256, 256, 0, stream>>>(dinv, N_NODES);
    degree_kernel<<<(N_EDGES + 255) / 256, 256, 0, stream>>>(dst, dinv, N_EDGES);
    dinv_kernel<<<(N_NODES + 255) / 256, 256, 0, stream>>>(dinv, N_NODES);
    norm_kernel<<<(N_EDGES + 255) / 256, 256, 0, stream>>>(src, dst, dinv, norm, N_EDGES);

    // --- 3 GCN layers ---
    const float* hin = x;
    int K = DIM_IN;
    const long aggElems = (long)N_NODES * DIM_H;
    for (int l = 0; l < 3; ++l) {
        gemm16_wmma<<<N_NODES / 16, 32 * (DIM_H / 16), 0, stream>>>(hin, W[l], hw, K, DIM_H);
        fill_zero<<<(int)((aggElems + 255) / 256), 256, 0, stream>>>(agg, aggElems);
        scatter_kernel<<<(int)(((long)N_EDGES * 16 + 255) / 256), 256, 0, stream>>>(
            hw, src, dst, norm, agg, N_EDGES);
        ln_relu_kernel<<<(int)(((long)N_NODES * 32 + 255) / 256), 256, 0, stream>>>(
            agg, b[l], g[l], t[l], h, N_NODES);
        hin = h;
        K = DIM_H;
    }

    // --- final linear (64 -> 40, padded to 3 column tiles) + log_softmax ---
    gemm16_wmma<<<N_NODES / 16, 32 * 3, 0, stream>>>(h, Wf, out, DIM_H, DIM_C);
    logsoftmax_kernel<<<(N_NODES + 255) / 256, 256, 0, stream>>>(out, bf, N_NODES);
}